// MarkovModel_16767552323887
// MI455X (gfx1250) — compile-verified
//
#include <hip/hip_runtime.h>
#include <cstdint>

#define B_DIM     64
#define S_DIM     4096
#define N_STATES  10000
#define TPB       256
#define T_PER_BLK 1024              // transitions handled per block
#define CHUNKS    4                 // ceil((S-1)/T_PER_BLK)
#define K_PER_THR (T_PER_BLK / TPB) // 4 independent gathers per thread

#define AS1 __attribute__((address_space(1)))
#define AS3 __attribute__((address_space(3)))

typedef int v4i __attribute__((vector_size(4 * sizeof(int))));
typedef AS1 v4i* g128_ptr;   // global src, b128 (matches builtin param 1)
typedef AS3 v4i* l128_ptr;   // LDS dst, b128
typedef AS1 int* g32_ptr;    // global src, b32
typedef AS3 int* l32_ptr;    // LDS dst, b32

// ---------------------------------------------------------------------------
// Kernel 1: one block per (sequence b, chunk c).
//   - Stage the contiguous seqs window [t0-1, t0+T_PER_BLK-1] into LDS via the
//     gfx1250 async global->LDS path (each element is read twice: as prev and
//     as cur state), wait on ASYNCcnt, barrier.
//   - Each thread issues K_PER_THR independent, predicated, NON-TEMPORAL
//     gathers into the 400MB transition table (no line reuse -> th:NT),
//     takes logs, accumulates.
//   - Deterministic LDS tree reduction -> partials[b*CHUNKS + c].
// ---------------------------------------------------------------------------
__global__ __launch_bounds__(TPB) void markov_partial_kernel(
    const int* __restrict__ seqs, const int* __restrict__ lengths,
    const float* __restrict__ trans, float* __restrict__ partials) {
  const int b   = blockIdx.x;
  const int c   = blockIdx.y;
  const int len = lengths[b];
  const int t0  = 1 + c * T_PER_BLK;

  __shared__ __align__(16) int s_seq[T_PER_BLK + 4];
  __shared__ float s_red[TPB];

  // Window base: seqs[b, t0-1]; (t0-1) is a multiple of 1024 -> 16B aligned.
  const int* g = seqs + (size_t)b * S_DIM + (t0 - 1);

#if defined(__gfx1250__) &&                                              \
    __has_builtin(__builtin_amdgcn_global_load_async_to_lds_b128) &&     \
    __has_builtin(__builtin_amdgcn_global_load_async_to_lds_b32)
  {
    // 256 lanes x 16B = 4KB = T_PER_BLK ints, async DMA into LDS.
    g128_ptr gsrc = (g128_ptr)(uintptr_t)(g + 4 * threadIdx.x);
    l128_ptr ldst = (l128_ptr)(uint32_t)(uintptr_t)(&s_seq[4 * threadIdx.x]);
    __builtin_amdgcn_global_load_async_to_lds_b128(gsrc, ldst, 0, 0);
    if (threadIdx.x == 0) {  // one tail element (x_{t0-1+T_PER_BLK})
      g32_ptr gs2 = (g32_ptr)(uintptr_t)(g + T_PER_BLK);
      l32_ptr ld2 = (l32_ptr)(uint32_t)(uintptr_t)(&s_seq[T_PER_BLK]);
      __builtin_amdgcn_global_load_async_to_lds_b32(gs2, ld2, 0, 0);
    }
#if __has_builtin(__builtin_amdgcn_s_wait_asynccnt)
    __builtin_amdgcn_s_wait_asynccnt(0);
#else
    asm volatile("s_wait_asynccnt 0" ::: "memory");
#endif
  }
#else
  for (int j = threadIdx.x; j <= T_PER_BLK; j += TPB) s_seq[j] = g[j];
#endif
  __syncthreads();

  // Compute all gather addresses first, then issue all loads (max MLP),
  // then the (latency-hidden) transcendental work.
  size_t ofs[K_PER_THR];
  bool   v[K_PER_THR];
#pragma unroll
  for (int k = 0; k < K_PER_THR; ++k) {
    const int j = threadIdx.x + k * TPB;   // local transition index
    const int t = t0 + j;                  // global target timestep
    v[k]   = (t < S_DIM) && (t < len);
    ofs[k] = (size_t)s_seq[j] * N_STATES + (size_t)s_seq[j + 1];
  }

  float p[K_PER_THR];
#pragma unroll
  for (int k = 0; k < K_PER_THR; ++k)
    p[k] = v[k] ? __builtin_nontemporal_load(trans + ofs[k]) : 1.0f;

  float acc = 0.0f;
#pragma unroll
  for (int k = 0; k < K_PER_THR; ++k)
    acc += logf(p[k]);                     // log(1)==0 for masked slots

  // Deterministic block tree reduction.
  s_red[threadIdx.x] = acc;
  __syncthreads();
#pragma unroll
  for (int s = TPB / 2; s > 0; s >>= 1) {
    if (threadIdx.x < (unsigned)s) s_red[threadIdx.x] += s_red[threadIdx.x + s];
    __syncthreads();
  }
  if (threadIdx.x == 0) partials[b * CHUNKS + c] = s_red[0];
}

// ---------------------------------------------------------------------------
// Kernel 2: single block, 64 threads (2 waves). Add log(init[x_0]), fold the
// chunk partials, then a serial (deterministic) 64-wide logsumexp; negate.
// ---------------------------------------------------------------------------
__global__ __launch_bounds__(B_DIM) void markov_finalize_kernel(
    const int* __restrict__ seqs, const float* __restrict__ init_probs,
    const float* __restrict__ partials, float* __restrict__ out) {
  __shared__ float s_ll[B_DIM];
  const int b = threadIdx.x;
  if (b < B_DIM) {
    float ll = logf(init_probs[seqs[(size_t)b * S_DIM]]);
#pragma unroll
    for (int c = 0; c < CHUNKS; ++c) ll += partials[b * CHUNKS + c];
    s_ll[b] = ll;
  }
  __syncthreads();
  if (threadIdx.x == 0) {
    float m = s_ll[0];
    for (int i = 1; i < B_DIM; ++i) m = fmaxf(m, s_ll[i]);
    float s = 0.0f;
    for (int i = 0; i < B_DIM; ++i) s += expf(s_ll[i] - m);
    out[0] = -(m + logf(s));
  }
}

// ---------------------------------------------------------------------------
extern "C" void kernel_launch(void* const* d_in, const int* in_sizes, int n_in,
                              void* d_out, int out_size, void* d_ws, size_t ws_size,
                              hipStream_t stream) {
  (void)in_sizes; (void)n_in; (void)out_size; (void)ws_size;
  const int*   seqs    = (const int*)d_in[0];     // [B, S]
  const int*   lengths = (const int*)d_in[1];     // [B]
  const float* init_p  = (const float*)d_in[2];   // [N_STATES]
  const float* trans_p = (const float*)d_in[3];   // [N_STATES, N_STATES]
  float*       out     = (float*)d_out;           // scalar
  float*       partials = (float*)d_ws;           // B*CHUNKS floats

  dim3 grid(B_DIM, CHUNKS);
  markov_partial_kernel<<<grid, TPB, 0, stream>>>(seqs, lengths, trans_p, partials);
  markov_finalize_kernel<<<1, B_DIM, 0, stream>>>(seqs, init_p, partials, out);
}